// OrthogonalButterflyLayerCUDA_40286793236491
// MI455X (gfx1250) — compile-verified
//
#include <hip/hip_runtime.h>

// Orthogonal butterfly, 32 stages, WIDTH=4096, BATCH=16384, fp32.
// Memory-bound: 512MB traffic -> ~22us floor @ 23.3 TB/s. Register-resident
// FFT-style phases (4 stages per phase in VGPRs), 7 swizzled-LDS exchanges,
// async global->LDS fill (ASYNCcnt), fully coalesced final store via the
// rotl8 output permutation.

#define WIDTH   4096
#define HALF    2048
#define DEPTH   32
#define BATCH   16384
#define THREADS 256
#define ROWS    8      // rows register-resident per block iteration
#define HROWS   4      // rows staged through the 64KB LDS scratch at a time
#define GRID    2048   // one 8-row iteration per block; grid-stride for safety

#ifndef USE_ASYNC_LDS
#if defined(__has_builtin)
#if __has_builtin(__builtin_amdgcn_global_load_async_to_lds_b128) && \
    __has_builtin(__builtin_amdgcn_s_wait_asynccnt)
#define USE_ASYNC_LDS 1
#endif
#endif
#endif
#ifndef USE_ASYNC_LDS
#define USE_ASYNC_LDS 0
#endif

#if USE_ASYNC_LDS
typedef int v4i __attribute__((ext_vector_type(4)));
typedef __attribute__((address_space(1))) v4i* glb_v4i_p;  // global (AS1)
typedef __attribute__((address_space(3))) v4i* lds_v4i_p;  // LDS (AS3)
#endif

// Word-index swizzle: XOR bits [4:2] with bits [8:6]. Keeps 16B alignment
// (bits [1:0] untouched) so b128 async copies stay contiguous; makes every
// exchange access pattern <=2-way bank conflicted.
__device__ __forceinline__ int swz(int l) { return l ^ (((l >> 6) & 7) << 2); }

// Logical element index for (thread u, register j) in a phase whose active
// bit-nibble base is N (N in {8,4,0}); u's 8 bits fill the remaining bits.
template <int N>
__device__ __forceinline__ int lidx(int u, int j) {
  if (N == 8) return (j << 8) | u;
  if (N == 4) return ((u >> 4) << 8) | (j << 4) | (u & 15);
  return (u << 4) | j;
}

template <int N>
__device__ __forceinline__ void lds_to_regs(float r[16], const float* buf, int u) {
#pragma unroll
  for (int j = 0; j < 16; ++j) r[j] = buf[swz(lidx<N>(u, j))];
}

template <int N>
__device__ __forceinline__ void regs_to_lds(const float r[16], float* buf, int u) {
#pragma unroll
  for (int j = 0; j < 16; ++j) buf[swz(lidx<N>(u, j))] = r[j];
}

// 4 butterfly stages with the active bit inside the register nibble N.
// Stage t (t0+tt) acts on bit b = N + m, m = 3-tt. Pair enumeration index
// i = compress(l0, b) matches the init kernel's table layout.
// The per-stage asm() launders the table pointer through an opaque SGPR so
// LICM cannot hoist all 256 table loads to the kernel top (which blew up
// live ranges and caused scratch spills); each stage's 8 loads stay local.
template <int N>
__device__ __forceinline__ void do_phase(float r[ROWS][16],
                                         const float2* tab, int t0, int u) {
#pragma unroll
  for (int tt = 0; tt < 4; ++tt) {
    const int m = 3 - tt;
    unsigned long long tv = (unsigned long long)(tab + (t0 + tt) * HALF);
    asm volatile("" : "+s"(tv));  // opaque: keeps these loads in this stage
    const float2* T = (const float2*)tv;
#pragma unroll
    for (int cj = 0; cj < 8; ++cj) {
      const int j0 = ((cj >> m) << (m + 1)) | (cj & ((1 << m) - 1));
      const int j1 = j0 | (1 << m);
      int i;
      if (N == 8)      i = (cj << 8) | u;
      else if (N == 4) i = ((u >> 4) << 7) | (cj << 4) | (u & 15);
      else             i = (u << 3) | cj;
      const float2 cs = T[i];  // shared across all ROWS rows
#pragma unroll
      for (int rr = 0; rr < ROWS; ++rr) {
        const float a = r[rr][j0];
        const float v = r[rr][j1];
        r[rr][j0] = fmaf(a, cs.x, v * cs.y);    // y0 =  c*x0 + s*x1
        r[rr][j1] = fmaf(v, cs.x, -(a * cs.y)); // y1 = -s*x0 + c*x1
      }
    }
  }
}

// Relayout registers from nibble NOLD to NNEW through the HROWS*WIDTH LDS
// scratch, ROWS/HROWS rows-groups at a time (LDS used only transiently).
template <int NOLD, int NNEW>
__device__ __forceinline__ void exchange(float r[ROWS][16], float* smem, int u) {
#pragma unroll
  for (int h = 0; h < ROWS / HROWS; ++h) {
#pragma unroll
    for (int q = 0; q < HROWS; ++q)
      regs_to_lds<NOLD>(r[h * HROWS + q], smem + q * WIDTH, u);
    __syncthreads();
#pragma unroll
    for (int q = 0; q < HROWS; ++q)
      lds_to_regs<NNEW>(r[h * HROWS + q], smem + q * WIDTH, u);
    __syncthreads();
  }
}

// Copy HROWS contiguous rows of X into swizzled LDS. Async path uses the
// CDNA5 global->LDS async instructions (tracked with ASYNCcnt).
__device__ __forceinline__ void fill_rows(const float* __restrict__ X,
                                          int row_base, float* smem, int u) {
#pragma unroll
  for (int q = 0; q < HROWS; ++q) {
    const float* src = X + (size_t)(row_base + q) * WIDTH;
    float* dstbase = smem + q * WIDTH;
#pragma unroll
    for (int c = 0; c < 4; ++c) {
      const int l4 = u * 4 + c * 1024;     // 16B-aligned word index
      float* dst = dstbase + swz(l4);      // swz preserves 16B alignment
#if USE_ASYNC_LDS
      __builtin_amdgcn_global_load_async_to_lds_b128(
          (glb_v4i_p)(unsigned long long)(const void*)(src + l4),
          (lds_v4i_p)(unsigned int)(unsigned long long)(void*)dst,
          0, 0);
#else
      *(float4*)dst = *(const float4*)(src + l4);
#endif
    }
  }
}

__device__ __forceinline__ void wait_async_zero() {
#if USE_ASYNC_LDS
  __builtin_amdgcn_s_wait_asynccnt(0);
#endif
}

// Main kernel first so the disasm snippet shows its prologue / fill path.
__global__ __launch_bounds__(THREADS) void butterfly_main(
    const float* __restrict__ X, const float2* __restrict__ tab,
    float* __restrict__ out) {
  __shared__ float smem[HROWS * WIDTH];  // 64 KB scratch
  const int u = threadIdx.x;
  float r[ROWS][16];

  for (int it = blockIdx.x; it < BATCH / ROWS; it += gridDim.x) {
    const int row_base = it * ROWS;

    // Load ROWS rows into registers in phase-A (N=8) layout, HROWS at a time.
#pragma unroll
    for (int h = 0; h < ROWS / HROWS; ++h) {
      fill_rows(X, row_base + h * HROWS, smem, u);
      wait_async_zero();
      __syncthreads();
#pragma unroll
      for (int q = 0; q < HROWS; ++q)
        lds_to_regs<8>(r[h * HROWS + q], smem + q * WIDTH, u);
      __syncthreads();
    }

    // Stage bit sequence: 11..8 | 7..4 | 3..0 | 11..8 | 7..4 | 3..0 | 11..8 | 7..4
    do_phase<8>(r, tab, 0, u);
    exchange<8, 4>(r, smem, u);
    do_phase<4>(r, tab, 4, u);
    exchange<4, 0>(r, smem, u);
    do_phase<0>(r, tab, 8, u);
    exchange<0, 8>(r, smem, u);
    do_phase<8>(r, tab, 12, u);
    exchange<8, 4>(r, smem, u);
    do_phase<4>(r, tab, 16, u);
    exchange<4, 0>(r, smem, u);
    do_phase<0>(r, tab, 20, u);
    exchange<0, 8>(r, smem, u);
    do_phase<8>(r, tab, 24, u);
    exchange<8, 4>(r, smem, u);
    do_phase<4>(r, tab, 28, u);

    // Output: out[p] = logical[rotr8(p)] <=> p = rotl8(l). In the N=4 layout
    // l = (u>>4)<<8 | j<<4 | (u&15), so p = (u&15)<<8 | (u>>4)<<4 | j:
    // each thread's 16 registers land on 16 CONTIGUOUS outputs.
    const int p_base = ((u & 15) << 8) | ((u >> 4) << 4);
#pragma unroll
    for (int rr = 0; rr < ROWS; ++rr) {
      float* dst = out + (size_t)(row_base + rr) * WIDTH + p_base;
#pragma unroll
      for (int j = 0; j < 16; j += 4) {
        *(float4*)(dst + j) =
            make_float4(r[rr][j], r[rr][j + 1], r[rr][j + 2], r[rr][j + 3]);
      }
    }
  }
}

// Init kernel: per (stage t, pair-enum i) precompute (cos,sin). Pair bit
// b = 11-(t%12); l0 = insert0(i,b); physical pair id k = rotl12^t(l0);
// theta = params[k, t] (params is (HALF, DEPTH) row-major).
__global__ void butterfly_init_tab(const float* __restrict__ params,
                                   float2* __restrict__ tab) {
  const int idx = blockIdx.x * blockDim.x + threadIdx.x;
  if (idx >= DEPTH * HALF) return;
  const int t = idx >> 11;
  const int i = idx & (HALF - 1);
  const int s = t % 12;
  const int b = 11 - s;
  const int l0 = ((i >> b) << (b + 1)) | (i & ((1 << b) - 1));
  const int k = ((l0 << s) | (l0 >> (12 - s))) & (WIDTH - 1);
  const float theta = params[k * DEPTH + t];
  float sv, cv;
  sincosf(theta, &sv, &cv);
  tab[idx] = make_float2(cv, sv);
}

extern "C" void kernel_launch(void* const* d_in, const int* in_sizes, int n_in,
                              void* d_out, int out_size, void* d_ws,
                              size_t ws_size, hipStream_t stream) {
  const float* X = (const float*)d_in[0];       // (16384, 4096) fp32
  const float* params = (const float*)d_in[1];  // (2048, 32) fp32
  float* out = (float*)d_out;                   // (16384, 4096) fp32
  float2* tab = (float2*)d_ws;                  // 32*2048 float2 = 512 KB

  butterfly_init_tab<<<(DEPTH * HALF + THREADS - 1) / THREADS, THREADS, 0,
                       stream>>>(params, tab);
  butterfly_main<<<GRID, THREADS, 0, stream>>>(X, tab, out);
}